// Pool_80822694576323
// MI455X (gfx1250) — compile-verified
//
#include <hip/hip_runtime.h>
#include <hip/hip_bf16.h>

typedef __attribute__((ext_vector_type(2))) float v2f;
typedef __attribute__((ext_vector_type(8))) float v8f;

#define RES   7
#define PTS   14          // RES * sample_ratio(2)
#define NPTS  196         // 14*14
#define CBLK  64          // channels per block (grid.y = 256/64)
#define SROW  68          // padded LDS row stride in floats

__global__ __launch_bounds__(256)
void roialign_wmma_kernel(const float* __restrict__ f0,
                          const float* __restrict__ f1,
                          const float* __restrict__ f2,
                          const float* __restrict__ f3,
                          const float* __restrict__ props0,
                          const float* __restrict__ props1,
                          float* __restrict__ out,
                          int n0 /* rois in props0 */)
{
    __shared__ float S[NPTS * SROW];          // bilinear samples [point][channel]
    __shared__ int   s_lo[2 * PTS];           // [0..13]=x axis, [14..27]=y axis
    __shared__ int   s_hi[2 * PTS];
    __shared__ float s_fr[2 * PTS];
    __shared__ float s_va[2 * PTS];

    const int roi = blockIdx.x;
    const int cg  = blockIdx.y;               // 64-channel group
    const int tid = threadIdx.x;

    // ---------------- ROI setup (wave-uniform scalar math) ----------------
    const float* box  = (roi < n0) ? (props0 + 4 * roi) : (props1 + 4 * (roi - n0));
    const int    bidx = (roi < n0) ? 0 : 1;
    const float x1 = box[0], y1 = box[1], x2 = box[2], y2 = box[3];

    const float area = (x2 - x1 + 1.0f) * (y2 - y1 + 1.0f);
    const float szr  = sqrtf(area);
    float lf = floorf(4.0f + log2f(szr / 224.0f + 1e-6f));
    lf = fminf(fmaxf(lf, 2.0f), 5.0f);
    const int lvl = (int)lf - 2;              // 0..3
    const float scale = 0.25f / (float)(1 << lvl);

    int H, W; const float* feat;
    if (lvl == 0)      { feat = f0; H = 200; W = 200; }
    else if (lvl == 1) { feat = f1; H = 100; W = 100; }
    else if (lvl == 2) { feat = f2; H = 50;  W = 50;  }
    else               { feat = f3; H = 25;  W = 25;  }

    const float x1s = x1 * scale, y1s = y1 * scale;
    const float x2s = x2 * scale, y2s = y2 * scale;
    const float roiw = fmaxf(x2s - x1s, 1.0f);
    const float roih = fmaxf(y2s - y1s, 1.0f);
    const float binw = roiw * (1.0f / RES);
    const float binh = roih * (1.0f / RES);

    // ---------------- Phase 0: per-axis interpolation tables ----------------
    if (tid < 2 * PTS) {
        const int axis = tid / PTS;           // 0 = x, 1 = y
        const int i    = tid % PTS;
        const float off = ((float)i + 0.5f) * 0.5f;     // (i+0.5)/sr, sr=2
        const float v   = axis ? (y1s + off * binh) : (x1s + off * binw);
        const int  size = axis ? H : W;
        const float va  = (v >= -1.0f && v <= (float)size) ? 1.0f : 0.0f;
        const float v0  = fmaxf(v, 0.0f);
        int lo = (int)floorf(v0);
        int hi; float fr;
        if (lo >= size - 1) { lo = size - 1; hi = size - 1; fr = 0.0f; }
        else                { hi = lo + 1;   fr = v0 - (float)lo; }
        s_lo[tid] = lo; s_hi[tid] = hi; s_fr[tid] = fr; s_va[tid] = va;
    }
    __syncthreads();

    // ---------------- Phase 1: gather bilinear samples into LDS ----------------
    {
        const int c_local = tid >> 2;                       // 0..63
        const int c       = cg * CBLK + c_local;
        const float* fc = feat + ((size_t)(bidx * 256 + c) * H) * W;
        #pragma unroll 1
        for (int p = (tid & 3); p < NPTS; p += 4) {         // lanes walk x-adjacent points
            const int sy = p / PTS, sx = p % PTS;
            const int   xlo = s_lo[sx],       xhi = s_hi[sx];
            const float fx  = s_fr[sx],       vx  = s_va[sx];
            const int   ylo = s_lo[PTS + sy], yhi = s_hi[PTS + sy];
            const float fy  = s_fr[PTS + sy], vy  = s_va[PTS + sy];
            const float hx = 1.0f - fx, hy = 1.0f - fy;
            const float* r0 = fc + (size_t)ylo * W;
            const float* r1 = fc + (size_t)yhi * W;
            const float v00 = r0[xlo], v01 = r0[xhi];
            const float v10 = r1[xlo], v11 = r1[xhi];
            const float val = hy * (hx * v00 + fx * v01) + fy * (hx * v10 + fx * v11);
            S[p * SROW + c_local] = val * (vx * vy);
        }
    }
    __syncthreads();

    // ---------------- Phase 2: pooled = Apool(49x196) * S(196x16ch) via WMMA ----------------
    const int wave = tid >> 5;                // 0..7
    const int lane = tid & 31;
    const int ln16 = lane & 15;
    const int hi16 = lane >> 4;               // 0 or 1
    const int ct   = wave & 3;                // channel tile within the 64-ch block
    const int mt0  = (wave >> 2) * 2;         // this wave handles m-tiles mt0, mt0+1
    const int cb   = ct * 16;

    for (int mi = 0; mi < 2; ++mi) {
        const int mt  = mt0 + mi;
        const int bin = mt * 16 + ln16;                 // A-matrix row (M) for this lane
        const bool binok = bin < RES * RES;
        const int by = bin / RES, bx = bin % RES;
        const int p00 = (by * 2) * PTS + bx * 2;        // the 4 samples feeding this bin:
                                                        // p00, p00+1, p00+PTS, p00+PTS+1
        v8f acc = {0.f, 0.f, 0.f, 0.f, 0.f, 0.f, 0.f, 0.f};

        for (int k = 0; k < 49; ++k) {                  // K = 196 in steps of 4
            const int pk = 4 * k;
            const int ka = pk + (hi16 ? 2 : 0);         // A layout: v0={K0|K2}, v1={K1|K3}
            v2f a, b;
            a.x = (binok && (ka     == p00 || ka     == p00 + 1 ||
                             ka     == p00 + PTS || ka == p00 + PTS + 1)) ? 0.25f : 0.0f;
            a.y = (binok && (ka + 1 == p00 || ka + 1 == p00 + 1 ||
                             ka + 1 == p00 + PTS || ka + 1 == p00 + PTS + 1)) ? 0.25f : 0.0f;
            // B layout [4x16]: row striped across lanes, v0={row0|row1}, v1={row2|row3}
            b.x = S[(pk     + hi16) * SROW + cb + ln16];
            b.y = S[(pk + 2 + hi16) * SROW + cb + ln16];
            acc = __builtin_amdgcn_wmma_f32_16x16x4_f32(
                      false, a, false, b, (short)0, acc, false, false);
        }

        // D layout: VGPR r -> M = r (lanes 0-15) / r+8 (lanes 16-31), N = ln16
        const int cc = cg * CBLK + cb + ln16;
        float* obase = out + ((size_t)roi * 256 + cc) * (RES * RES);
        #pragma unroll
        for (int r = 0; r < 8; ++r) {
            const int obin = mt * 16 + r + hi16 * 8;
            if (obin < RES * RES) obase[obin] = acc[r];
        }
    }
}

extern "C" void kernel_launch(void* const* d_in, const int* in_sizes, int n_in,
                              void* d_out, int out_size, void* d_ws, size_t ws_size,
                              hipStream_t stream) {
    const float* f0 = (const float*)d_in[0];
    const float* f1 = (const float*)d_in[1];
    const float* f2 = (const float*)d_in[2];
    const float* f3 = (const float*)d_in[3];
    // d_in[4] (feat4) is unused by the reference (only 4 ratios/levels).
    const float* p0 = (const float*)d_in[5];
    const float* p1 = (const float*)d_in[6];
    float* out = (float*)d_out;

    const int n0 = in_sizes[5] / 4;
    const int n1 = in_sizes[6] / 4;
    const int R  = n0 + n1;                    // 1024 ROIs

    dim3 grid(R, 256 / CBLK);                  // (1024, 4)
    roialign_wmma_kernel<<<grid, 256, 0, stream>>>(f0, f1, f2, f3, p0, p1, out, n0);
}